// HerbEncoder_7069516169928
// MI455X (gfx1250) — compile-verified
//
#include <hip/hip_runtime.h>
#include <math.h>

typedef __attribute__((ext_vector_type(2))) float v2f;
typedef __attribute__((ext_vector_type(8))) float v8f;
typedef __attribute__((ext_vector_type(4))) unsigned u32x4;
typedef __attribute__((ext_vector_type(8))) int i32x8;
typedef __attribute__((ext_vector_type(4))) int i32x4;

#define HID 128
#define NHEAD 8

// ---------------------------------------------------------------------------
// Tensor Data Mover: DMA a 2D f32 tile (tileCols x tileRows, row stride
// strideElems) from global memory into LDS. Issued by one wave; completion
// via TENSORcnt. Descriptor per CDNA5 ISA ch.8 (D# groups).
// ---------------------------------------------------------------------------
__device__ __forceinline__ void tdm_load_2d_f32(void* ldsDst, const void* src,
                                                unsigned tileCols, unsigned tileRows,
                                                unsigned strideElems) {
  unsigned lds_off = (unsigned)(unsigned long long)(uintptr_t)ldsDst; // low 32b = LDS offset
  unsigned long long ga = (unsigned long long)(uintptr_t)src;
  u32x4 g0;
  g0[0] = 1u;                                            // count=1, user descriptor
  g0[1] = lds_off;                                       // lds_addr [63:32]
  g0[2] = (unsigned)(ga & 0xFFFFFFFFull);                // global_addr lo
  g0[3] = (unsigned)((ga >> 32) & 0x01FFFFFFull)         // global_addr hi (57-bit)
        | (2u << 30);                                    // type=2 ("image")
  i32x8 g1;
  g1[0] = (int)(2u << 16);                               // data_size=2 -> 4 bytes
  g1[1] = (int)((tileCols & 0xFFFFu) << 16);             // tensor_dim0 lo16 @bits[63:48]
  g1[2] = (int)(((tileCols >> 16) & 0xFFFFu)             // tensor_dim0 hi16
        | ((tileRows & 0xFFFFu) << 16));                 // tensor_dim1 lo16
  g1[3] = (int)(((tileRows >> 16) & 0xFFFFu)             // tensor_dim1 hi16
        | ((tileCols & 0xFFFFu) << 16));                 // tile_dim0 @bits[127:112]
  g1[4] = (int)(tileRows & 0xFFFFu);                     // tile_dim1 (tile_dim2=0)
  g1[5] = (int)strideElems;                              // tensor_dim0_stride lo32
  g1[6] = 0;                                             // dim0_stride hi / dim1_stride lo
  g1[7] = 0;
  i32x4 gz = {0, 0, 0, 0};                               // 2D tensor: groups 2/3 unused
#if __has_include(<hip/amd_detail/amd_gfx1250_TDM.h>)
  i32x8 gz8 = {0, 0, 0, 0, 0, 0, 0, 0};
  __builtin_amdgcn_tensor_load_to_lds(g0, g1, gz, gz, gz8, 0);
#else
  __builtin_amdgcn_tensor_load_to_lds(g0, g1, gz, gz, 0);
#endif
}

// Stage a K x 16 column tile of B into Bs via TDM (wave 0) + barrier.
__device__ __forceinline__ void stage_B_tile(float* Bs, const float* Bcol,
                                             unsigned rows, unsigned strideElems) {
  if (threadIdx.x < 32) {                    // wave 0 issues the DMA
    tdm_load_2d_f32(Bs, Bcol, 16u, rows, strideElems);
    __builtin_amdgcn_s_wait_tensorcnt(0);    // wave-local completion
  }
  __syncthreads();                           // publish LDS to all waves
}

// ---------------------------------------------------------------------------
// WMMA f32 16x16x4 accumulate loop with A in global memory, B staged in LDS
// as Bs[k][16] (conflict-free: lanes 0-15 -> banks 16k+l, 16-31 -> 16k+32+l).
// A layout: lane l<16 holds M=l, a.x=A[M,k+2*half], a.y=A[M,k+2*half+1]
// B layout: lane l<16 holds N=l, b.x=Bs[k+2*half][N], b.y=Bs[k+2*half+1][N]
// ---------------------------------------------------------------------------
__device__ __forceinline__ v8f wmma_loop_lds(const float* __restrict__ A, int lda,
                                             const float* Bs, int K, v8f acc) {
  const int lane = threadIdx.x & 31;
  const int l16  = lane & 15;
  const int half = lane >> 4;
  const float* ap = A + (size_t)l16 * lda + 2 * half;
  const float* bp = Bs + (2 * half) * 16 + l16;
#pragma unroll 4
  for (int k = 0; k < K; k += 4) {
    float2 av = *(const float2*)(ap + k);
    v2f a; a.x = av.x; a.y = av.y;
    v2f b; b.x = bp[k * 16]; b.y = bp[k * 16 + 16];
    acc = __builtin_amdgcn_wmma_f32_16x16x4_f32(false, a, false, b,
                                                (short)0, acc, false, false);
  }
  return acc;
}

// C/D layout: VGPR r: lanes 0-15 -> M=r, lanes 16-31 -> M=8+r; col = l16
template <bool HAS_BIAS>
__device__ __forceinline__ void store_tile(float* __restrict__ C, int ldc,
                                           const float* __restrict__ bias, v8f acc) {
  const int lane = threadIdx.x & 31;
  const int l16  = lane & 15;
  const int half = lane >> 4;
  const float b = HAS_BIAS ? bias[l16] : 0.0f;
#pragma unroll
  for (int r = 0; r < 8; ++r) {
    float v = HAS_BIAS ? (acc[r] + b) : acc[r];
    C[(size_t)(half * 8 + r) * ldc + l16] = v;
  }
}

// ---------------------------------------------------------------------------
// Block-cooperative GEMM + bias: C[M,N] = A[M,K] @ B[K,N] + bias[N], N=128.
// Block = 8 waves sharing one 16-wide N tile of B (TDM-staged in LDS); each
// wave owns one 16-row M tile. Grid = (ceil(M/16/8), N/16).
// ---------------------------------------------------------------------------
template <int K>
__global__ __launch_bounds__(256) void gemm_bias_k(
    const float* __restrict__ A, const float* __restrict__ B,
    const float* __restrict__ bias, float* __restrict__ C,
    int M, int N) {
  __shared__ float Bs[K * 16];
  const int tN = blockIdx.y;
  stage_B_tile(Bs, B + tN * 16, (unsigned)K, (unsigned)N);
  int tM = blockIdx.x * 8 + (threadIdx.x >> 5);
  if (tM * 16 >= M) return;              // wave-uniform; EXEC all-1 for WMMA
  v8f acc = {};
  acc = wmma_loop_lds(A + (size_t)tM * 16 * K, K, Bs, K, acc);
  store_tile<true>(C + (size_t)tM * 16 * N + tN * 16, N, bias + tN * 16, acc);
}

// ---------------------------------------------------------------------------
// Relation transform: out[n, h*16+e] = sum_d src[n, h*16+d] * R[h, d, e].
// Block = 8 waves = one 16-node tile x 8 heads; all 8 head matrices (8 KB,
// contiguous -> one TDM tile of 128 rows x 16 cols) staged in LDS.
// ---------------------------------------------------------------------------
__global__ __launch_bounds__(256) void rel_k(
    const float* __restrict__ S, const float* __restrict__ R,
    float* __restrict__ O) {
  __shared__ float Rs[NHEAD * 256];
  stage_B_tile(Rs, R, 128u, 16u);        // contiguous 2048 floats
  const int tM = blockIdx.x;
  const int h  = threadIdx.x >> 5;
  v8f acc = {};
  acc = wmma_loop_lds(S + (size_t)tM * 16 * HID + h * 16, HID, Rs + h * 256, 16, acc);
  store_tile<false>(O + (size_t)tM * 16 * HID + h * 16, HID, nullptr, acc);
}

// ---------------------------------------------------------------------------
// Edge pass 1: logits + segment max (monotone uint encoding for atomicMax)
// ---------------------------------------------------------------------------
__device__ __forceinline__ unsigned enc_f(float f) {
  unsigned u = __float_as_uint(f);
  return (u & 0x80000000u) ? ~u : (u | 0x80000000u);
}
__device__ __forceinline__ float dec_f(unsigned u) {
  u = (u & 0x80000000u) ? (u & 0x7FFFFFFFu) : ~u;
  return __uint_as_float(u);
}

__global__ __launch_bounds__(256) void edge_logit_max_k(
    const int* __restrict__ src, const int* __restrict__ dst,
    const float* __restrict__ q,     // global [68000,128]
    const float* __restrict__ krel,  // pre-offset to this edge type's rows
    const float* __restrict__ prel,  // [8] for this edge type
    float* __restrict__ logits,      // pre-offset, [E,8]
    unsigned* __restrict__ mx,       // global [68000,8]
    int E, int dstOff) {
  int e = blockIdx.x * blockDim.x + threadIdx.x;
  if (e >= E) return;
  int s = src[e], d = dst[e];
  const float4* qv = (const float4*)(q + (size_t)(dstOff + d) * HID);
  const float4* kv = (const float4*)(krel + (size_t)s * HID);
#pragma unroll
  for (int h = 0; h < NHEAD; ++h) {
    float acc = 0.0f;
#pragma unroll
    for (int j = 0; j < 4; ++j) {
      float4 qq = qv[h * 4 + j], kk = kv[h * 4 + j];
      acc += qq.x * kk.x + qq.y * kk.y + qq.z * kk.z + qq.w * kk.w;
    }
    float lg = acc * prel[h] * 0.25f;   // p_rel / sqrt(D), D=16
    logits[(size_t)e * NHEAD + h] = lg;
    atomicMax(&mx[(size_t)(dstOff + d) * NHEAD + h], enc_f(lg));
  }
}

// ---------------------------------------------------------------------------
// Edge pass 2: ex = exp(logit - mx); den += ex; agg += ex * v_rel[src]
// One thread per (edge, head).
// ---------------------------------------------------------------------------
__global__ __launch_bounds__(256) void edge_exp_agg_k(
    const int* __restrict__ src, const int* __restrict__ dst,
    const float* __restrict__ vrel,     // pre-offset
    const float* __restrict__ logits,   // pre-offset
    const unsigned* __restrict__ mx,
    float* __restrict__ den, float* __restrict__ agg,
    int E, int dstOff) {
  int idx = blockIdx.x * blockDim.x + threadIdx.x;
  if (idx >= E * NHEAD) return;
  int e = idx >> 3, h = idx & 7;
  int s = src[e], d = dst[e];
  float m  = dec_f(mx[(size_t)(dstOff + d) * NHEAD + h]);
  float ex = expf(logits[(size_t)e * NHEAD + h] - m);
  atomicAdd(&den[(size_t)(dstOff + d) * NHEAD + h], ex);
  const float4* vv = (const float4*)(vrel + (size_t)s * HID + h * 16);
  float* ag = agg + (size_t)(dstOff + d) * HID + h * 16;
#pragma unroll
  for (int j = 0; j < 4; ++j) {
    float4 v = vv[j];
    atomicAdd(ag + j * 4 + 0, ex * v.x);
    atomicAdd(ag + j * 4 + 1, ex * v.y);
    atomicAdd(ag + j * 4 + 2, ex * v.z);
    atomicAdd(ag + j * 4 + 3, ex * v.w);
  }
}

// ---------------------------------------------------------------------------
// Normalize by denominator + exact GELU, in place over agg[68000,128]
// ---------------------------------------------------------------------------
__global__ __launch_bounds__(256) void norm_gelu_k(
    float* __restrict__ agg, const float* __restrict__ den, int total) {
  int i = blockIdx.x * blockDim.x + threadIdx.x;
  if (i >= total) return;
  int n = i >> 7, c = i & 127, h = c >> 4;
  float v = agg[i] / (den[(size_t)n * NHEAD + h] + 1e-16f);
  agg[i] = 0.5f * v * (1.0f + erff(v * 0.70710678118654752f));
}

// ---------------------------------------------------------------------------
// GEMM (K=N=128) + bias + sigmoid-skip-gate + ReLU epilogue:
//   C = relu(g*(A@B+bias) + (1-g)*xs),  g = sigmoid(skip[0])
// Block-cooperative, TDM-staged B like gemm_bias_k.
// ---------------------------------------------------------------------------
__global__ __launch_bounds__(256) void gemm_skip_k(
    const float* __restrict__ A, const float* __restrict__ B,
    const float* __restrict__ bias, const float* __restrict__ xs,
    const float* __restrict__ skipp, float* __restrict__ C, int M) {
  __shared__ float Bs[HID * 16];
  const int tN = blockIdx.y;
  stage_B_tile(Bs, B + tN * 16, (unsigned)HID, (unsigned)HID);
  int tM = blockIdx.x * 8 + (threadIdx.x >> 5);
  if (tM * 16 >= M) return;
  v8f acc = {};
  acc = wmma_loop_lds(A + (size_t)tM * 16 * HID, HID, Bs, HID, acc);
  const int lane = threadIdx.x & 31, l16 = lane & 15, half = lane >> 4;
  float g = 1.0f / (1.0f + expf(-skipp[0]));
  float b = bias[tN * 16 + l16];
  const float* xp = xs + (size_t)tM * 16 * HID + tN * 16;
  float* cp = C + (size_t)tM * 16 * HID + tN * 16;
#pragma unroll
  for (int r = 0; r < 8; ++r) {
    int row = half * 8 + r;
    float o = acc[r] + b;
    float val = g * o + (1.0f - g) * xp[(size_t)row * HID + l16];
    cp[(size_t)row * HID + l16] = fmaxf(val, 0.0f);
  }
}

// ---------------------------------------------------------------------------
// Host launcher
// ---------------------------------------------------------------------------
extern "C" void kernel_launch(void* const* d_in, const int* in_sizes, int n_in,
                              void* d_out, int out_size, void* d_ws, size_t ws_size,
                              hipStream_t stream) {
  const int NH = 8000, NI = 40000, NT = 20000, NTOT = 68000;
  const int OFF_H = 0, OFF_I = 8000, OFF_T = 48000;

  const float* x_in[3] = {(const float*)d_in[0], (const float*)d_in[1], (const float*)d_in[2]};
  const float* W_in[3] = {(const float*)d_in[3], (const float*)d_in[5], (const float*)d_in[7]};
  const float* b_in[3] = {(const float*)d_in[4], (const float*)d_in[6], (const float*)d_in[8]};
  const float* Wk = (const float*)d_in[9];   const float* bk = (const float*)d_in[10];
  const float* Wq = (const float*)d_in[11];  const float* bq = (const float*)d_in[12];
  const float* Wv = (const float*)d_in[13];  const float* bv = (const float*)d_in[14];
  const float* a_rel = (const float*)d_in[15];
  const float* m_rel = (const float*)d_in[16];
  const float* p_rel = (const float*)d_in[17];
  const float* Wa = (const float*)d_in[18];  const float* ba = (const float*)d_in[19];
  const float* skip = (const float*)d_in[20];
  const float* W_out = (const float*)d_in[21];
  const float* b_out = (const float*)d_in[22];

  const int* esrc[4] = {(const int*)d_in[23], (const int*)d_in[25],
                        (const int*)d_in[27], (const int*)d_in[29]};
  const int* edst[4] = {(const int*)d_in[24], (const int*)d_in[26],
                        (const int*)d_in[28], (const int*)d_in[30]};
  const int  E[4]    = {in_sizes[23], in_sizes[25], in_sizes[27], in_sizes[29]};
  const int  srcTypeOff[4] = {OFF_H, OFF_I, OFF_I, OFF_T};   // rows of k/v buffers
  const int  srcCnt[4]     = {NH, NI, NI, NT};
  const int  krelOff[4]    = {0, 8000, 48000, 88000};        // rows in krel/vrel
  const int  dstOff[4]     = {OFF_I, OFF_H, OFF_T, OFF_I};

  // ---- workspace layout (floats) ----
  float* ws = (float*)d_ws;
  size_t o = 0;
  float* xs   = ws + o; o += (size_t)NTOT * HID;
  float* kbuf = ws + o; o += (size_t)NTOT * HID;
  float* qbuf = ws + o; o += (size_t)NTOT * HID;
  float* vbuf = ws + o; o += (size_t)NTOT * HID;
  float* krel = ws + o; o += (size_t)108000 * HID;
  float* vrel = ws + o; o += (size_t)108000 * HID;
  float* logit = ws + o;
  size_t lgOff[4]; size_t etot = 0;
  for (int t = 0; t < 4; ++t) { lgOff[t] = etot * NHEAD; etot += (size_t)E[t]; }
  o += etot * NHEAD;
  unsigned* mx = (unsigned*)(ws + o); size_t zeroStart = o; o += (size_t)NTOT * NHEAD;
  float* den  = ws + o; o += (size_t)NTOT * NHEAD;
  float* agg  = ws + o; o += (size_t)NTOT * HID;
  float* hherb = ws + o; o += (size_t)NH * HID;
  size_t zeroCount = ((size_t)NTOT * NHEAD * 2 + (size_t)NTOT * HID) * sizeof(float);
  (void)ws_size; (void)n_in; (void)out_size;

  float* out = (float*)d_out;
  float* out_slice[3] = {out, out + (size_t)NH * HID, out + (size_t)(NH + NI) * HID};
  const int Ntype[3] = {NH, NI, NT};
  const int typeOff[3] = {OFF_H, OFF_I, OFF_T};

  auto cdiv = [](long long a, long long b) { return (int)((a + b - 1) / b); };
  auto grid2 = [&](int M) { return dim3((unsigned)cdiv(M >> 4, 8), HID >> 4, 1); };

  // 1) input projections -> xs  (K = 400 / 300 / 200, compile-time)
  gemm_bias_k<400><<<grid2(NH), 256, 0, stream>>>(
      x_in[0], W_in[0], b_in[0], xs + (size_t)OFF_H * HID, NH, HID);
  gemm_bias_k<300><<<grid2(NI), 256, 0, stream>>>(
      x_in[1], W_in[1], b_in[1], xs + (size_t)OFF_I * HID, NI, HID);
  gemm_bias_k<200><<<grid2(NT), 256, 0, stream>>>(
      x_in[2], W_in[2], b_in[2], xs + (size_t)OFF_T * HID, NT, HID);

  // 2) k/q/v per node type (K = 128)
  for (int i = 0; i < 3; ++i) {
    const float* xt = xs + (size_t)typeOff[i] * HID;
    gemm_bias_k<HID><<<grid2(Ntype[i]), 256, 0, stream>>>(
        xt, Wk + (size_t)i * HID * HID, bk + (size_t)i * HID,
        kbuf + (size_t)typeOff[i] * HID, Ntype[i], HID);
    gemm_bias_k<HID><<<grid2(Ntype[i]), 256, 0, stream>>>(
        xt, Wq + (size_t)i * HID * HID, bq + (size_t)i * HID,
        qbuf + (size_t)typeOff[i] * HID, Ntype[i], HID);
    gemm_bias_k<HID><<<grid2(Ntype[i]), 256, 0, stream>>>(
        xt, Wv + (size_t)i * HID * HID, bv + (size_t)i * HID,
        vbuf + (size_t)typeOff[i] * HID, Ntype[i], HID);
  }
  // 3) relation transforms at node level (k_rel, v_rel per edge type)
  for (int t = 0; t < 4; ++t) {
    int mt = srcCnt[t] >> 4;
    rel_k<<<mt, 256, 0, stream>>>(
        kbuf + (size_t)srcTypeOff[t] * HID, a_rel + (size_t)t * NHEAD * 256,
        krel + (size_t)krelOff[t] * HID);
    rel_k<<<mt, 256, 0, stream>>>(
        vbuf + (size_t)srcTypeOff[t] * HID, m_rel + (size_t)t * NHEAD * 256,
        vrel + (size_t)krelOff[t] * HID);
  }
  // 4) zero mx / den / agg (contiguous region)
  hipMemsetAsync(ws + zeroStart, 0, zeroCount, stream);

  // 5) edge pass 1: logits + segment max
  for (int t = 0; t < 4; ++t) {
    edge_logit_max_k<<<cdiv(E[t], 256), 256, 0, stream>>>(
        esrc[t], edst[t], qbuf, krel + (size_t)krelOff[t] * HID,
        p_rel + (size_t)t * NHEAD, logit + lgOff[t], mx, E[t], dstOff[t]);
  }
  // 6) edge pass 2: exp + denominator + message aggregation
  for (int t = 0; t < 4; ++t) {
    edge_exp_agg_k<<<cdiv((long long)E[t] * NHEAD, 256), 256, 0, stream>>>(
        esrc[t], edst[t], vrel + (size_t)krelOff[t] * HID,
        logit + lgOff[t], mx, den, agg, E[t], dstOff[t]);
  }
  // 7) normalize + GELU (in place on agg)
  {
    long long total = (long long)NTOT * HID;
    norm_gelu_k<<<cdiv(total, 256), 256, 0, stream>>>(agg, den, (int)total);
  }
  // 8) output linear + skip gate + relu per type (herb -> scratch, ing/tgt -> d_out)
  for (int i = 0; i < 3; ++i) {
    float* dstp = (i == 0) ? hherb : out_slice[i];
    gemm_skip_k<<<grid2(Ntype[i]), 256, 0, stream>>>(
        agg + (size_t)typeOff[i] * HID, Wa + (size_t)i * HID * HID,
        ba + (size_t)i * HID, xs + (size_t)typeOff[i] * HID,
        skip + i, dstp, Ntype[i]);
  }
  // 9) final herb linear -> d_out[0 : NH*HID]
  gemm_bias_k<HID><<<grid2(NH), 256, 0, stream>>>(
      hherb, W_out, b_out, out_slice[0], NH, HID);
}